// GCNMLP_15960098472302
// MI455X (gfx1250) — compile-verified
//
#include <hip/hip_runtime.h>
#include <hip/hip_bf16.h>

#define BATCH 32
#define EPB   2048
#define DIN   64
#define EMB   128
#define HID   64
#define PRED  3
#define NPRED 12
#define NNODE (BATCH * EPB)          /* 65536 */
#define NEDGE (NNODE * 16)           /* 1048576 */

typedef __attribute__((ext_vector_type(2))) float v2f;
typedef __attribute__((ext_vector_type(8))) float v8f;

__device__ __forceinline__ v8f wmma_f32_k4(v2f a, v2f b, v8f c) {
    // D = A(16x4 f32) * B(4x16 f32) + C(16x16 f32)
    return __builtin_amdgcn_wmma_f32_16x16x4_f32(
        /*neg_a=*/false, a, /*neg_b=*/false, b,
        /*c_mod=*/(short)0, c, /*reuse_a=*/false, /*reuse_b=*/false);
}

__device__ __forceinline__ void atomic_add_f32(float* p, float v) {
    unsafeAtomicAdd(p, v);   // lowers to global_atomic_add_f32
}

// Stage W[KH,128] into LDS as K-pairs: sW2[(k/2)*128+n] = {W[k][n], W[k+1][n]}
// so each WMMA B fragment is a single aligned ds_load_b64.
template <int KH>
__device__ __forceinline__ void stage_weights_paired(const float* __restrict__ W,
                                                     float* __restrict__ sW) {
    // KH/2 pair-rows x (EMB/4) float4 column groups
    for (int i = threadIdx.x; i < (KH / 2) * (EMB / 4); i += 256) {
        int p  = i / (EMB / 4);
        int n4 = (i % (EMB / 4)) * 4;
        float4 r0 = *reinterpret_cast<const float4*>(W + (2 * p) * EMB + n4);
        float4 r1 = *reinterpret_cast<const float4*>(W + (2 * p + 1) * EMB + n4);
        float2* dst = reinterpret_cast<float2*>(sW) + (size_t)p * EMB + n4;
        dst[0] = make_float2(r0.x, r1.x);
        dst[1] = make_float2(r0.y, r1.y);
        dst[2] = make_float2(r0.z, r1.z);
        dst[3] = make_float2(r0.w, r1.w);
    }
}

// One K-sweep: acc[2][8] += A(32 rows) x Wstaged, A row stride = KH
template <int KH>
__device__ __forceinline__ void wmma_sweep(const float* __restrict__ Src, int rowbase,
                                           int m16, int koff, const float* __restrict__ sW,
                                           v8f acc0[8], v8f acc1[8]) {
    const float* arow0 = Src + (size_t)(rowbase + m16) * KH;
    const float* arow1 = Src + (size_t)(rowbase + 16 + m16) * KH;
    for (int k = 0; k < KH; k += 4) {
        v2f a0 = *reinterpret_cast<const v2f*>(arow0 + k + koff);
        v2f a1 = *reinterpret_cast<const v2f*>(arow1 + k + koff);
        const v2f* brow = reinterpret_cast<const v2f*>(sW) + (size_t)((k + koff) >> 1) * EMB;
        #pragma unroll
        for (int nt = 0; nt < 8; ++nt) {
            v2f b = brow[nt * 16 + m16];
            acc0[nt] = wmma_f32_k4(a0, b, acc0[nt]);
            acc1[nt] = wmma_f32_k4(a1, b, acc1[nt]);
        }
    }
}

// ---------------------------------------------------------------- utilities
__global__ __launch_bounds__(256) void zero_kernel(float* p, int n) {
    int i = blockIdx.x * 256 + threadIdx.x;
    if (i < n) p[i] = 0.0f;
}

__global__ __launch_bounds__(256) void degree_kernel(const long long* __restrict__ ei,
                                                     float* __restrict__ deg) {
    int e = blockIdx.x * 256 + threadIdx.x;
    if (e < NEDGE) {
        long long s = ei[e];               // src row of edge_index
        atomic_add_f32(&deg[s], 1.0f);
    }
}

__global__ __launch_bounds__(256) void dinv_kernel(float* __restrict__ deg) {
    int i = blockIdx.x * 256 + threadIdx.x;
    if (i < NNODE) {
        float d = deg[i];
        deg[i] = (d > 0.0f) ? rsqrtf(d) : 0.0f;   // in-place -> dinv
    }
}

// ----------------------------------------------------- edge gather/scatter
// one wave per edge: tx[dst,:] += (-dinv[src]*dinv[dst]) * h[src,:]
__global__ __launch_bounds__(256) void scatter_kernel(const long long* __restrict__ ei,
                                                      const float* __restrict__ dinv,
                                                      const float* __restrict__ H,
                                                      float* __restrict__ Tx) {
    int wave = blockIdx.x * 8 + (threadIdx.x >> 5);
    int lane = threadIdx.x & 31;
    if (wave >= NEDGE) return;
    long long s = ei[wave];
    long long d = ei[NEDGE + wave];
    float w = -dinv[s] * dinv[d];
    const float4 v = *reinterpret_cast<const float4*>(H + (size_t)s * EMB + lane * 4);
    float* out = Tx + (size_t)d * EMB + lane * 4;
    atomic_add_f32(out + 0, w * v.x);
    atomic_add_f32(out + 1, w * v.y);
    atomic_add_f32(out + 2, w * v.z);
    atomic_add_f32(out + 3, w * v.w);
}

// ------------------------------------------------------------- embed GEMM
// H[N,128] = X[N,64] @ W[64,128] + b ; wave = 32 rows x 128 cols (16 WMMA tiles)
__global__ __launch_bounds__(256) void embed_gemm(const float* __restrict__ X,
                                                  const float* __restrict__ W,
                                                  const float* __restrict__ bias,
                                                  float* __restrict__ H) {
    __shared__ float sW[DIN * EMB];                     // 32 KB, K-paired
    stage_weights_paired<DIN>(W, sW);
    __syncthreads();

    const int lane = threadIdx.x & 31;
    const int wid  = threadIdx.x >> 5;
    const int rowbase = blockIdx.x * 256 + wid * 32;
    const int m16  = lane & 15;
    const int koff = (lane >> 4) * 2;

    v8f acc0[8] = {}, acc1[8] = {};
    wmma_sweep<DIN>(X, rowbase, m16, koff, sW, acc0, acc1);

    const int mh = (lane >> 4) * 8;
    #pragma unroll
    for (int nt = 0; nt < 8; ++nt) {
        float bv = bias[nt * 16 + m16];
        #pragma unroll
        for (int r = 0; r < 8; ++r) {
            H[(size_t)(rowbase + mh + r) * EMB + nt * 16 + m16]      = acc0[nt][r] + bv;
            H[(size_t)(rowbase + 16 + mh + r) * EMB + nt * 16 + m16] = acc1[nt][r] + bv;
        }
    }
}

// ------------------------------------------------------------- conv GEMM
// Hout = relu(Hin @ W0 + Tx @ W1 + b), all [N,128] x [128,128]
__global__ __launch_bounds__(256) void conv_gemm(const float* __restrict__ Hin,
                                                 const float* __restrict__ Tx,
                                                 const float* __restrict__ W0,
                                                 const float* __restrict__ W1,
                                                 const float* __restrict__ bias,
                                                 float* __restrict__ Hout) {
    __shared__ float sW[EMB * EMB];                     // 64 KB, staged twice, K-paired
    const int lane = threadIdx.x & 31;
    const int wid  = threadIdx.x >> 5;
    const int rowbase = blockIdx.x * 256 + wid * 32;
    const int m16  = lane & 15;
    const int koff = (lane >> 4) * 2;

    v8f acc0[8] = {}, acc1[8] = {};

    // ---- pass 0: Hin @ W0
    stage_weights_paired<EMB>(W0, sW);
    __syncthreads();
    wmma_sweep<EMB>(Hin, rowbase, m16, koff, sW, acc0, acc1);
    __syncthreads();

    // ---- pass 1: Tx @ W1 (accumulate)
    stage_weights_paired<EMB>(W1, sW);
    __syncthreads();
    wmma_sweep<EMB>(Tx, rowbase, m16, koff, sW, acc0, acc1);

    const int mh = (lane >> 4) * 8;
    #pragma unroll
    for (int nt = 0; nt < 8; ++nt) {
        float bv = bias[nt * 16 + m16];
        #pragma unroll
        for (int r = 0; r < 8; ++r) {
            float v0 = acc0[nt][r] + bv;
            float v1 = acc1[nt][r] + bv;
            Hout[(size_t)(rowbase + mh + r) * EMB + nt * 16 + m16]      = fmaxf(v0, 0.0f);
            Hout[(size_t)(rowbase + 16 + mh + r) * EMB + nt * 16 + m16] = fmaxf(v1, 0.0f);
        }
    }
}

// --------------------------------------------------------------- pooling
__global__ __launch_bounds__(128) void pool_kernel(const float* __restrict__ H,
                                                   float* __restrict__ pooled) {
    int b = blockIdx.x, j = threadIdx.x;
    const float* base = H + (size_t)b * EPB * EMB + j;
    float s = 0.0f;
    for (int e = 0; e < EPB; ++e) s += base[(size_t)e * EMB];
    pooled[b * EMB + j] = s * (1.0f / (float)EPB);
}

// ------------------------------------------------------------------- MLP
__global__ __launch_bounds__(256) void mlp_kernel(const float* __restrict__ pooled,
                                                  const float* __restrict__ W1,
                                                  const float* __restrict__ b1,
                                                  const float* __restrict__ W2,
                                                  const float* __restrict__ b2,
                                                  float* __restrict__ out1) {
    __shared__ float sp[BATCH * EMB];
    __shared__ float sh[BATCH * HID];
    for (int i = threadIdx.x; i < BATCH * EMB; i += 256) sp[i] = pooled[i];
    __syncthreads();
    for (int i = threadIdx.x; i < BATCH * HID; i += 256) {
        int b = i / HID, j = i % HID;
        float acc = b1[j];
        for (int k = 0; k < EMB; ++k) acc += sp[b * EMB + k] * W1[k * HID + j];
        sh[i] = fmaxf(acc, 0.0f);
    }
    __syncthreads();
    for (int i = threadIdx.x; i < BATCH * PRED; i += 256) {
        int b = i / PRED, c = i % PRED;
        float acc = b2[c];
        for (int k = 0; k < HID; ++k) acc += sh[b * HID + k] * W2[k * PRED + c];
        out1[i] = acc;
    }
}

// ------------------------------------------------------------- broadcast
__global__ __launch_bounds__(256) void bcast_kernel(const float* __restrict__ out1,
                                                    float* __restrict__ out, int n) {
    int idx = blockIdx.x * 256 + threadIdx.x;
    if (idx < n) {
        int c = idx % PRED;
        int b = idx / (NPRED * EPB * PRED);
        out[idx] = out1[b * PRED + c];
    }
}

extern "C" void kernel_launch(void* const* d_in, const int* in_sizes, int n_in,
                              void* d_out, int out_size, void* d_ws, size_t ws_size,
                              hipStream_t stream) {
    const float*     x       = (const float*)d_in[0];
    const long long* ei      = (const long long*)d_in[1];
    const float*     embedW  = (const float*)d_in[2];
    const float*     embedB  = (const float*)d_in[3];
    const float*     c0W0    = (const float*)d_in[4];
    const float*     c0W1    = (const float*)d_in[5];
    const float*     c0b     = (const float*)d_in[6];
    const float*     c1W0    = (const float*)d_in[7];
    const float*     c1W1    = (const float*)d_in[8];
    const float*     c1b     = (const float*)d_in[9];
    const float*     mW1     = (const float*)d_in[10];
    const float*     mb1     = (const float*)d_in[11];
    const float*     mW2     = (const float*)d_in[12];
    const float*     mb2     = (const float*)d_in[13];

    float* ws = (float*)d_ws;
    const size_t HSZ = (size_t)NNODE * EMB;            // 8,388,608 floats
    float* hA     = ws;
    float* hB     = ws + HSZ;
    float* tx     = ws + 2 * HSZ;
    float* deg    = ws + 3 * HSZ;                      // becomes dinv in place
    float* pooled = deg + NNODE;
    float* out1   = pooled + BATCH * EMB;

    const int gemm_grid = NNODE / 256;                 // 256 blocks, 8 waves x 32 rows

    // degrees / normalization
    zero_kernel<<<NNODE / 256, 256, 0, stream>>>(deg, NNODE);
    degree_kernel<<<NEDGE / 256, 256, 0, stream>>>(ei, deg);
    dinv_kernel<<<NNODE / 256, 256, 0, stream>>>(deg);

    // embed
    embed_gemm<<<gemm_grid, 256, 0, stream>>>(x, embedW, embedB, hA);

    // conv0
    zero_kernel<<<(int)(HSZ / 256), 256, 0, stream>>>(tx, (int)HSZ);
    scatter_kernel<<<NEDGE / 8, 256, 0, stream>>>(ei, deg, hA, tx);
    conv_gemm<<<gemm_grid, 256, 0, stream>>>(hA, tx, c0W0, c0W1, c0b, hB);

    // conv1
    zero_kernel<<<(int)(HSZ / 256), 256, 0, stream>>>(tx, (int)HSZ);
    scatter_kernel<<<NEDGE / 8, 256, 0, stream>>>(ei, deg, hB, tx);
    conv_gemm<<<gemm_grid, 256, 0, stream>>>(hB, tx, c1W0, c1W1, c1b, hA);

    // pool + MLP + broadcast
    pool_kernel<<<BATCH, 128, 0, stream>>>(hA, pooled);
    mlp_kernel<<<1, 256, 0, stream>>>(pooled, mW1, mb1, mW2, mb2, out1);
    bcast_kernel<<<(out_size + 255) / 256, 256, 0, stream>>>(out1, (float*)d_out, out_size);
}